// SparseLinear_35175782154586
// MI455X (gfx1250) — compile-verified
//
#include <hip/hip_runtime.h>
#include <hip/hip_bf16.h>

typedef float v2f __attribute__((ext_vector_type(2)));
typedef float v4f __attribute__((ext_vector_type(4)));
typedef float v8f __attribute__((ext_vector_type(8)));
typedef int   v4i __attribute__((ext_vector_type(4)));

#define IN 128
#define OUT 128
#define ROWS 32           // rows of x per tile iteration (two 16-row WMMA tiles)
#define LDS_STRIDE 132    // padded floats per row: bank-conflict-free A-frag reads
#define THREADS 256       // 8 wave32 waves -> 8 x 16 output columns = 128

#ifndef __has_builtin
#define __has_builtin(x) 0
#endif

#if __has_builtin(__builtin_amdgcn_global_load_async_to_lds_b128)
#define USE_ASYNC_LDS 1
#else
#define USE_ASYNC_LDS 0
#endif

#define AS1 __attribute__((address_space(1)))
#define AS3 __attribute__((address_space(3)))

// Issue (or perform, on non-async toolchains) the copy of one 32x128 f32 tile
// of x into an LDS buffer. 256 threads x 4 x b128 = 16 KB, fully coalesced.
__device__ __forceinline__ void stage_tile(const float* __restrict__ xin,
                                           float* __restrict__ ldsbuf, int tid)
{
#pragma unroll
    for (int p = 0; p < 4; ++p) {
        const int idx = p * THREADS + tid;     // float4 index 0..1023
        const int r   = idx >> 5;              // row 0..31
        const int c4  = (idx & 31) << 2;       // col 0..124
#if USE_ASYNC_LDS
        __builtin_amdgcn_global_load_async_to_lds_b128(
            (AS1 v4i*)(v4i*)(void*)(xin + r * IN + c4),
            (AS3 v4i*)(v4i*)(void*)(ldsbuf + r * LDS_STRIDE + c4),
            0, 0);
#else
        const v4f v = *(const v4f*)(xin + r * IN + c4);
        *(v4f*)(ldsbuf + r * LDS_STRIDE + c4) = v;
#endif
    }
}

__device__ __forceinline__ void wait_prev_tile()
{
#if USE_ASYNC_LDS
#if __has_builtin(__builtin_amdgcn_s_wait_asynccnt)
    __builtin_amdgcn_s_wait_asynccnt(4);   // 4 newest still in flight; older done
#else
    asm volatile("s_wait_asynccnt 4" ::: "memory");
#endif
#endif
}

__global__ __launch_bounds__(THREADS)
void sparse_linear_wmma_f32(const float* __restrict__ x,
                            const float* __restrict__ w,
                            const float* __restrict__ m,
                            const float* __restrict__ bias,
                            float* __restrict__ out,
                            int num_tiles)
{
    __shared__ float lds[2][ROWS * LDS_STRIDE];   // double buffer, 33.8 KB

    const int tid   = threadIdx.x;
    const int lane  = tid & 31;
    const int wave  = tid >> 5;        // 0..7
    const int c0    = wave * 16;       // this wave's output-column block
    const int mrow  = lane & 15;       // M (or N) index within 16
    const int khalf = lane >> 4;       // 0: K pair {4j,4j+1}; 1: K pair {4j+2,4j+3}

    // ---------------- Preload B fragments into registers ----------------
    // B is (W*mask)^T restricted to cols [c0, c0+16): B[k][n] = W'[c0+n][k].
    // 16x4-f32 B-frag j: v0 = rows K=4j (lanes 0-15) / 4j+2 (lanes 16-31),
    //                    v1 = rows K=4j+1 / 4j+3. Lane l holds column n=l%16.
    v2f bfrag[32];
#pragma unroll
    for (int j = 0; j < 32; ++j) {
        const int o = c0 + mrow;              // output channel = W row
        const int k = 4 * j + 2 * khalf;      // K column (even -> 8B aligned)
        const v2f wv = *(const v2f*)(w + o * IN + k);
        const v2f mv = *(const v2f*)(m + o * IN + k);
        bfrag[j] = wv * mv;
    }

    const float bval = bias[c0 + mrow];       // C/D layout: lane holds N = lane%16

    // ---------------- Software pipeline over row tiles ----------------
    stage_tile(x + (size_t)blockIdx.x * ROWS * IN, &lds[0][0], tid);

    int i = 0;
    for (int t = blockIdx.x; t < num_tiles; t += gridDim.x, ++i) {
        const int cur = i & 1;
        const int tn  = t + (int)gridDim.x;
        const int tp  = (tn < num_tiles) ? tn : t;     // uniform clamp (EXEC all-1)

        // Prefetch next tile into the other buffer; its target was freed by the
        // post-compute barrier of iteration i-1.
        stage_tile(x + (size_t)tp * ROWS * IN, &lds[cur ^ 1][0], tid);

        wait_prev_tile();      // only waits for tile t's 4 loads (in-order done)
        __syncthreads();       // tile t visible to all 8 waves

        const float* buf = &lds[cur][0];

        // -------- WMMA over K = 128 (32 x K4 steps), two 16-row tiles --------
        v8f acc0, acc1;
#pragma unroll
        for (int ii = 0; ii < 8; ++ii) { acc0[ii] = bval; acc1[ii] = bval; }

#pragma unroll
        for (int j = 0; j < 32; ++j) {
            const int k = 4 * j + 2 * khalf;
            // 16x4-f32 A-frag: lane l holds x[row = l%16][k .. k+1]
            // (ds_load_2addr_b64, conflict-free with stride 132)
            const v2f a0 = *(const v2f*)(buf + mrow * LDS_STRIDE + k);
            const v2f a1 = *(const v2f*)(buf + (mrow + 16) * LDS_STRIDE + k);
            acc0 = __builtin_amdgcn_wmma_f32_16x16x4_f32(
                false, a0, false, bfrag[j], (short)0, acc0, false, false);
            acc1 = __builtin_amdgcn_wmma_f32_16x16x4_f32(
                false, a1, false, bfrag[j], (short)0, acc1, false, false);
        }

        __syncthreads();       // done reading lds[cur]; frees it for iter i+1's prefetch target two steps ahead

        // -------- Store D: VGPR ii -> row (ii + 8*khalf), col c0 + lane%16 ----
        const size_t rowBase = (size_t)t * ROWS;
        float* o0 = out + (rowBase + (size_t)(8 * khalf)) * OUT + c0 + mrow;
        float* o1 = o0 + (size_t)16 * OUT;
#pragma unroll
        for (int ii = 0; ii < 8; ++ii) {
            o0[(size_t)ii * OUT] = acc0[ii];
            o1[(size_t)ii * OUT] = acc1[ii];
        }
    }
}

extern "C" void kernel_launch(void* const* d_in, const int* in_sizes, int n_in,
                              void* d_out, int out_size, void* d_ws, size_t ws_size,
                              hipStream_t stream) {
    const float* x    = (const float*)d_in[0];
    const float* wgt  = (const float*)d_in[1];
    const float* mask = (const float*)d_in[2];
    const float* bias = (const float*)d_in[3];
    float* out = (float*)d_out;

    const int rows = in_sizes[0] / IN;        // 1048576
    const int num_tiles = rows / ROWS;        // 32768
    int blocks = num_tiles < 2048 ? num_tiles : 2048;

    hipLaunchKernelGGL(sparse_linear_wmma_f32, dim3(blocks), dim3(THREADS), 0, stream,
                       x, wgt, mask, bias, out, num_tiles);
}